// SecondOrderFeatureInteraction_28930899706274
// MI455X (gfx1250) — compile-verified
//
#include <hip/hip_runtime.h>

typedef __attribute__((ext_vector_type(2))) float v2f;
typedef __attribute__((ext_vector_type(8))) float v8f;

// One wave32 computes one batch's 64x64 Gram matrix G = X * X^T (X is 64x32 f32)
// via V_WMMA_F32_16X16X4_F32, and scatters the strict upper triangle (2016 vals).
//
// f32 16x4 A-tile layout (ISA 7.12.2): lane L holds row (L&15), cols {c, c+1}
// with c = 2*(L>>4). The 4x16 B-tile of X^T has identical per-lane content
// (lane = N, VGPR/half = K), so one set of registers serves as both A and B.
//
// Diagonal-tile masking uses the same-wave store-ordering guarantee: invalid
// lanes store to dummy slot ob[15] (element (0,16) of tile (0,1)), which is
// rewritten with its correct value by the off-diagonal pass issued later in
// program order. Stores from one wave to the same address stay ordered, so
// the final value is the correct one — no EXEC-mask manipulation needed.
__global__ __launch_bounds__(256) void gram_triu_kernel(const float* __restrict__ X,
                                                        float* __restrict__ out,
                                                        int nbatch) {
  const int lane = (int)(threadIdx.x & 31);
  // Wave-uniform batch index pinned to an SGPR -> saddr-form memory ops.
  const int b = __builtin_amdgcn_readfirstlane((int)blockIdx.x * 8 +
                                               (int)(threadIdx.x >> 5));
  if (b >= nbatch) return;  // wave-uniform exit; EXEC stays all-1s for WMMA

  const float* __restrict__ xb = X + (size_t)b * (64 * 32);
  const int r  = lane & 15;          // row within 16-row block / N within tile
  const int ch = (lane >> 4) << 1;   // column base: 0 (lanes 0-15) or 2 (16-31)
  const int mh = (lane >> 4) << 3;   // M offset within C/D tile: 0 or 8
  const int t  = r - mh;             // diag predicate threshold: store iff v < t

  // Load all of X for this batch in WMMA A/B layout: 4 blocks x 8 k-slices.
  v2f a[4][8];
#pragma unroll
  for (int i = 0; i < 4; ++i) {
    const int rowoff = (i * 16 + r) * 32 + ch;
#pragma unroll
    for (int s = 0; s < 8; ++s) {
      a[i][s] = *(const v2f*)(xb + rowoff + 4 * s);  // global_load_b64 saddr
    }
  }

  float* __restrict__ ob = out + (size_t)b * 2016;

  // ---- Pass 1: diagonal tiles (i,i); invalid lanes dump to ob[15]. ----
#pragma unroll
  for (int i = 0; i < 4; ++i) {
    v8f acc = {0.f, 0.f, 0.f, 0.f, 0.f, 0.f, 0.f, 0.f};
#pragma unroll
    for (int s = 0; s < 8; ++s) {
      acc = __builtin_amdgcn_wmma_f32_16x16x4_f32(
          false, a[i][s], false, a[i][s], (short)0, acc, false, false);
    }
    // Lane owns (m = 16i + mh + v, n = 16i + r); triu offset incrementally:
    //   off(m,n) = 63m - m(m-1)/2 + (n - m - 1);  off(v+1)-off(v) = 62 - m
    const int n = i * 16 + r;
    int m = i * 16 + mh;
    int off = m * 63 - ((m * (m - 1)) >> 1) + (n - m - 1);
#pragma unroll
    for (int v = 0; v < 8; ++v) {
      const int soff = (v < t) ? off : 15;  // v_cndmask, no EXEC manipulation
      ob[soff] = acc[v];
      off += 62 - m;
      ++m;
    }
  }

  // ---- Pass 2: off-diagonal tiles (i<j); fully upper, unconditional. ----
  // Tile (0,1) is first: its v=0 store rewrites ob[15] with the true value.
#pragma unroll
  for (int i = 0; i < 4; ++i) {
#pragma unroll
    for (int j = i + 1; j < 4; ++j) {
      v8f acc = {0.f, 0.f, 0.f, 0.f, 0.f, 0.f, 0.f, 0.f};
#pragma unroll
      for (int s = 0; s < 8; ++s) {
        acc = __builtin_amdgcn_wmma_f32_16x16x4_f32(
            false, a[i][s], false, a[j][s], (short)0, acc, false, false);
      }
      const int n = j * 16 + r;
      int m = i * 16 + mh;
      int off = m * 63 - ((m * (m - 1)) >> 1) + (n - m - 1);
#pragma unroll
      for (int v = 0; v < 8; ++v) {
        ob[off] = acc[v];
        off += 62 - m;
        ++m;
      }
    }
  }
}

extern "C" void kernel_launch(void* const* d_in, const int* in_sizes, int n_in,
                              void* d_out, int out_size, void* d_ws, size_t ws_size,
                              hipStream_t stream) {
  const float* X = (const float*)d_in[0];
  float* out = (float*)d_out;
  const int nbatch = in_sizes[0] / (64 * 32);  // 8192 for the reference shapes
  const int nblocks = (nbatch + 7) / 8;        // 8 waves (batches) per 256-thr block
  gram_triu_kernel<<<dim3(nblocks), dim3(256), 0, stream>>>(X, out, nbatch);
}